// lct_64596308132224
// MI455X (gfx1250) — compile-verified
//
#include <hip/hip_runtime.h>

// ---------------------------------------------------------------------------
// LCT NLOS reconstruction for gfx1250 (MI455X).
//   - Sequential per-(b,d) pipeline: the 134MB padded complex volume stays
//     resident in the 192MB L2 across all FFT passes.
//   - Matmuls: V_WMMA_F32_16X16X4_F32 (f32 in, f32 acc).
//   - FFT: 256-pt Stockham radix-2 in LDS. Contiguous axis -> 1 line/block;
//     strided axes -> 8-line tiles so every global access is a contiguous
//     64B run (vs 8B gathers).
//   - Fusions: zero-pad folded into fwd pass 1; Wiener filter folded into
//     fwd pass 3 store; ifft 1/N^3 scale + real-crop folded into final GEMM.
// ---------------------------------------------------------------------------

typedef float v2f __attribute__((ext_vector_type(2)));
typedef float v8f __attribute__((ext_vector_type(8)));

#define TCROP   128
#define NSP     128
#define NPAD    256
#define HWN     (NSP * NSP)               // 16384
#define VOXELS  (TCROP * NSP * NSP)       // 2097152  (128^3)
#define PADVOX  (NPAD * NPAD * NPAD)      // 16777216 (256^3)
#define TWOPI   6.2831853071795864f

#define TLINES  8       // lines per tiled-FFT block
#define LSTRIDE 258     // padded per-line LDS stride (float2)

// ---------------------------------------------------------------------------
// 1) Time placement (tbes offset) + gridz^4 falloff.
// ---------------------------------------------------------------------------
__global__ void __launch_bounds__(256)
prep_kernel(const float* __restrict__ feat,   // (T,H,W) slice for this (b,d)
            const float* __restrict__ gridz,  // (T)
            const int*   __restrict__ tbes,
            int b,
            float* __restrict__ outv)         // (T,H,W)
{
    const int i = blockIdx.x * 256 + threadIdx.x;       // < 128^3
    const int w = i & 127;
    const int h = (i >> 7) & 127;
    const int t = i >> 14;
    const int ts = t - tbes[b];
    const float g  = gridz[t];
    const float g2 = g * g;
    const float g4 = g2 * g2;
    float v = 0.0f;
    if (ts >= 0 && ts < TCROP)
        v = feat[(ts << 14) + (h << 7) + w] * g4;
    outv[i] = v;
}

// ---------------------------------------------------------------------------
// 2) C(128 x 16384) = A(128x128) * B(128x16384) via V_WMMA_F32_16X16X4_F32.
//    One wave per 16x16 tile, 8 waves/block over adjacent n-tiles.
// ---------------------------------------------------------------------------
__global__ void __launch_bounds__(256)
gemm_mt_wmma(const float* __restrict__ Amt,   // (128,128) row-major (m,t)
             const float* __restrict__ Bsrc,  // (t, 16384)
             float* __restrict__ Cdst)        // (m, 16384)
{
    const int lane = threadIdx.x & 31;
    const int wv   = threadIdx.x >> 5;
    const int n0   = (blockIdx.x * 8 + wv) * 16;
    const int m0   = blockIdx.y * 16;
    const int half = lane >> 4;
    const int lid  = lane & 15;

    v8f acc = {};
    for (int k = 0; k < TCROP; k += 4) {
        const int ka = k + (half << 1);
        v2f a, b;
        a.x = Amt[(m0 + lid) * TCROP + ka];
        a.y = Amt[(m0 + lid) * TCROP + ka + 1];
        b.x = Bsrc[ka * HWN + n0 + lid];
        b.y = Bsrc[(ka + 1) * HWN + n0 + lid];
        acc = __builtin_amdgcn_wmma_f32_16x16x4_f32(
                  false, a, false, b, (short)0, acc, false, false);
    }
#pragma unroll
    for (int r = 0; r < 8; ++r)
        Cdst[(m0 + r + half * 8) * HWN + n0 + lid] = acc[r];
}

// ---------------------------------------------------------------------------
// Stockham radix-2 butterfly stage body (shared by all FFT kernels).
// ---------------------------------------------------------------------------
__device__ __forceinline__ void stockham_bfly(
    const float2* __restrict__ A, float2* __restrict__ Bq,
    int ofs, int j, int st, float sign)
{
    const int s  = 1 << st;
    const int nn = NPAD >> st;
    const int mh = nn >> 1;
    const int q  = j & (s - 1);
    const int p  = j >> st;

    const float2 a = A[ofs + q + s * p];
    const float2 b = A[ofs + q + s * (p + mh)];
    float wc, ws;
    const float ang = sign * TWOPI * (float)p / (float)nn;
    __sincosf(ang, &ws, &wc);

    float2 sum, rot;
    sum.x = a.x + b.x;  sum.y = a.y + b.y;
    const float dx = a.x - b.x, dy = a.y - b.y;
    rot.x = dx * wc - dy * ws;
    rot.y = dx * ws + dy * wc;

    Bq[ofs + q + s * (2 * p)]     = sum;
    Bq[ofs + q + s * (2 * p + 1)] = rot;
}

// ---------------------------------------------------------------------------
// 3) Forward pass 1 (w axis, stride 1) fused with zero-padding:
//    reads real 128^3 (zero outside), writes complex 256^3.
// ---------------------------------------------------------------------------
__global__ void __launch_bounds__(128)
fft256_first(const float* __restrict__ src,   // 128^3 real
             float2* __restrict__ dst)        // 256^3 complex
{
    __shared__ float2 sa[NPAD];
    __shared__ float2 sb[NPAD];

    const int t    = threadIdx.x;             // 0..127
    const int line = blockIdx.x;              // m*256 + h
    const int m    = line >> 8;
    const int h    = line & 255;
    const int base = line * NPAD;             // m*65536 + h*256

    float2 v0 = make_float2(0.0f, 0.0f);
    if (m < TCROP && h < NSP)
        v0.x = src[(m << 14) + (h << 7) + t];
    sa[t]       = v0;
    sa[t + 128] = make_float2(0.0f, 0.0f);
    __syncthreads();

    float2* A = sa; float2* Bq = sb;
#pragma unroll
    for (int st = 0; st < 8; ++st) {
        stockham_bfly(A, Bq, 0, t, st, -1.0f);
        __syncthreads();
        float2* tp = A; A = Bq; Bq = tp;
    }

    dst[base + t]       = A[t];
    dst[base + t + 128] = A[t + 128];
}

// ---------------------------------------------------------------------------
// 4) Inverse pass 3 (w axis, stride 1): one line per 128-thread block.
// ---------------------------------------------------------------------------
__global__ void __launch_bounds__(128)
fft256_w_inv(float2* __restrict__ buf)
{
    __shared__ float2 sa[NPAD];
    __shared__ float2 sb[NPAD];

    const int t    = threadIdx.x;
    const int base = blockIdx.x * NPAD;

    sa[t]       = buf[base + t];
    sa[t + 128] = buf[base + t + 128];
    __syncthreads();

    float2* A = sa; float2* Bq = sb;
#pragma unroll
    for (int st = 0; st < 8; ++st) {
        stockham_bfly(A, Bq, 0, t, st, 1.0f);
        __syncthreads();
        float2* tp = A; A = Bq; Bq = tp;
    }

    buf[base + t]       = A[t];
    buf[base + t + 128] = A[t + 128];
}

// ---------------------------------------------------------------------------
// 5) Tiled FFT pass for strided axes (h: stride 256, m: stride 65536).
//    8 adjacent-w lines per 256-thread block: global accesses are contiguous
//    64B runs; each wave32 owns one line in LDS for the butterfly stages.
//    Optional fused complex filter multiply at store time (pr/pi non-null).
//    base = blockIdx.y * slowStride + blockIdx.x * TLINES; grid (32, 256).
// ---------------------------------------------------------------------------
__global__ void __launch_bounds__(256)
fft256_tiled(float2* __restrict__ buf, int elemStride, int slowStride,
             float sign,
             const float* __restrict__ pr, const float* __restrict__ pi)
{
    __shared__ float2 sa[TLINES * LSTRIDE];
    __shared__ float2 sb[TLINES * LSTRIDE];

    const int tid  = threadIdx.x;
    const int base = blockIdx.y * slowStride + blockIdx.x * TLINES;

    // coalesced load: 8 cols (w) x 256 rows (fft axis), 64B runs
#pragma unroll
    for (int k = 0; k < 8; ++k) {
        const int e   = k * 256 + tid;
        const int col = e & (TLINES - 1);
        const int row = e >> 3;
        sa[col * LSTRIDE + row] = buf[base + row * elemStride + col];
    }
    __syncthreads();

    const int line = tid >> 5;       // wave id: one line per wave32
    const int wk   = tid & 31;
    const int ofs  = line * LSTRIDE;

    float2* A = sa; float2* Bq = sb;
#pragma unroll
    for (int st = 0; st < 8; ++st) {
#pragma unroll
        for (int i = 0; i < 4; ++i)
            stockham_bfly(A, Bq, ofs, wk + 32 * i, st, sign);
        __syncthreads();
        float2* tp = A; A = Bq; Bq = tp;
    }

    // coalesced store, optionally fused with Wiener filter multiply
#pragma unroll
    for (int k = 0; k < 8; ++k) {
        const int e   = k * 256 + tid;
        const int col = e & (TLINES - 1);
        const int row = e >> 3;
        const int gi  = base + row * elemStride + col;
        float2 v = A[col * LSTRIDE + row];
        if (pr != nullptr) {
            const float fr = pr[gi], fi = pi[gi];
            v = make_float2(v.x * fr - v.y * fi, v.x * fi + v.y * fr);
        }
        buf[gi] = v;
    }
}

// ---------------------------------------------------------------------------
// 6) out = mtxi * Re(crop(vol)) * (1/N^3), WMMA f32.
// ---------------------------------------------------------------------------
__global__ void __launch_bounds__(256)
gemm_mti_wmma(const float* __restrict__ Amt,        // mtxi (m,t) row-major
              const float2* __restrict__ vol,       // 256^3 complex
              float* __restrict__ outp,             // (m, 16384)
              float scale)
{
    const int lane = threadIdx.x & 31;
    const int wv   = threadIdx.x >> 5;
    const int n0   = (blockIdx.x * 8 + wv) * 16;
    const int m0   = blockIdx.y * 16;
    const int half = lane >> 4;
    const int lid  = lane & 15;

    const int n  = n0 + lid;
    const int h  = n >> 7;
    const int w  = n & 127;
    const int sp = h * NPAD + w;

    v8f acc = {};
    for (int k = 0; k < TCROP; k += 4) {
        const int ka = k + (half << 1);
        v2f a, b;
        a.x = Amt[(m0 + lid) * TCROP + ka];
        a.y = Amt[(m0 + lid) * TCROP + ka + 1];
        b.x = vol[ka * (NPAD * NPAD) + sp].x * scale;
        b.y = vol[(ka + 1) * (NPAD * NPAD) + sp].x * scale;
        acc = __builtin_amdgcn_wmma_f32_16x16x4_f32(
                  false, a, false, b, (short)0, acc, false, false);
    }
#pragma unroll
    for (int r = 0; r < 8; ++r)
        outp[(m0 + r + half * 8) * HWN + n] = acc[r];
}

// ---------------------------------------------------------------------------
// Launch: sequential over the 4 (b,d) slices, single shared complex volume.
// Workspace: [0, 134MB) complex vol; then two 8MB real temps.  ~151MB total.
// ---------------------------------------------------------------------------
extern "C" void kernel_launch(void* const* d_in, const int* in_sizes, int n_in,
                              void* d_out, int out_size, void* d_ws, size_t ws_size,
                              hipStream_t stream)
{
    (void)in_sizes; (void)n_in; (void)out_size; (void)ws_size;

    const float* feat  = (const float*)d_in[0];
    const float* gridz = (const float*)d_in[1];
    const float* mtx   = (const float*)d_in[2];
    const float* mtxi  = (const float*)d_in[3];
    const float* pr    = (const float*)d_in[4];
    const float* pi    = (const float*)d_in[5];
    const int*   tbes  = (const int*)d_in[6];
    float*       outp  = (float*)d_out;

    float2* cplx = (float2*)d_ws;
    float*  tmp0 = (float*)((char*)d_ws + (size_t)PADVOX * sizeof(float2));
    float*  tmp1 = tmp0 + VOXELS;

    const dim3 gemmGrid(HWN / (16 * 8), TCROP / 16);   // (128, 8)
    const dim3 tileGrid(NPAD / TLINES, NPAD);          // (32, 256)
    const float invScale = 1.0f / (float)PADVOX;

    for (int bd = 0; bd < 4; ++bd) {
        const int b = bd >> 1;
        const float* fbd = feat + (size_t)bd * VOXELS;

        prep_kernel<<<VOXELS / 256, 256, 0, stream>>>(fbd, gridz, tbes, b, tmp0);
        gemm_mt_wmma<<<gemmGrid, 256, 0, stream>>>(mtx, tmp0, tmp1);

        // forward 3D FFT (pad fused into pass 1; filter fused into pass 3)
        fft256_first<<<NPAD * NPAD, 128, 0, stream>>>(tmp1, cplx);
        fft256_tiled<<<tileGrid, 256, 0, stream>>>(cplx, NPAD, NPAD * NPAD,
                                                   -1.0f, nullptr, nullptr);
        fft256_tiled<<<tileGrid, 256, 0, stream>>>(cplx, NPAD * NPAD, NPAD,
                                                   -1.0f, pr, pi);

        // inverse 3D FFT (unscaled; 1/N^3 folded into final GEMM)
        fft256_tiled<<<tileGrid, 256, 0, stream>>>(cplx, NPAD * NPAD, NPAD,
                                                   1.0f, nullptr, nullptr);
        fft256_tiled<<<tileGrid, 256, 0, stream>>>(cplx, NPAD, NPAD * NPAD,
                                                   1.0f, nullptr, nullptr);
        fft256_w_inv<<<NPAD * NPAD, 128, 0, stream>>>(cplx);

        gemm_mti_wmma<<<gemmGrid, 256, 0, stream>>>(
            mtxi, cplx, outp + (size_t)bd * VOXELS, invScale);
    }
}